// MethylSPWNet_52656299049592
// MI455X (gfx1250) — compile-verified
//
#include <hip/hip_runtime.h>
#include <hip/hip_bf16.h>

// ---------------------------------------------------------------------------
// MethylSPWNet forward for MI455X (gfx1250, wave32).
//   Stage 1: memory-bound segmented scatter-reduce (64x800000 fp32 ~205MB,
//            HBM floor ~9us @ 23.3TB/s) -> streaming b128 loads + sorted-idx
//            running sums + LDS f32 atomics + per-block global f32 atomics.
//   Stage 2: tiny fused head on ONE workgroup using V_WMMA_F32_16X16X4_F32
//            (fp32-exact CDNA5 matrix path) for the three small GEMMs.
// ---------------------------------------------------------------------------

#define B_   64
#define P_   128
#define H1_  256
#define H2_  128
#define C_   32

typedef __attribute__((ext_vector_type(2))) float v2f;
typedef __attribute__((ext_vector_type(8))) float v8f;

// ---------------------------------------------------------------------------
// Stage 1: WX[b,p] += x[b,i]*w[i] for idx[i]==p. idx is sorted.
// Each thread owns 64 contiguous features (N=800000 is a multiple of 64).
// ---------------------------------------------------------------------------
__global__ void __launch_bounds__(256) spw_scatter(
    const float* __restrict__ x, const float* __restrict__ w,
    const int* __restrict__ idx, float* __restrict__ WX, int N)
{
    __shared__ float lacc[P_];
    const int t = threadIdx.x;
    if (t < P_) lacc[t] = 0.0f;
    __syncthreads();

    const int  b    = blockIdx.y;
    const long base = ((long)blockIdx.x * 256 + t) * 64;

    if (base < N) {
        const float* xr = x + (long)b * N + base;
        const float* wr = w + base;
        const int*   ir = idx + base;
        int   cur = ir[0];
        float run = 0.0f;
        #pragma unroll 4
        for (int j = 0; j < 64; j += 4) {
            float4 xv = *(const float4*)(xr + j);
            float4 wv = *(const float4*)(wr + j);
            int4   iv = *(const int4*)(ir + j);
            float p0 = xv.x * wv.x, p1 = xv.y * wv.y;
            float p2 = xv.z * wv.z, p3 = xv.w * wv.w;
            if (iv.x != cur) { atomicAdd(&lacc[cur], run); run = 0.f; cur = iv.x; } run += p0;
            if (iv.y != cur) { atomicAdd(&lacc[cur], run); run = 0.f; cur = iv.y; } run += p1;
            if (iv.z != cur) { atomicAdd(&lacc[cur], run); run = 0.f; cur = iv.z; } run += p2;
            if (iv.w != cur) { atomicAdd(&lacc[cur], run); run = 0.f; cur = iv.w; } run += p3;
        }
        atomicAdd(&lacc[cur], run);
    }
    __syncthreads();

    if (t < P_) {
        float v = lacc[t];
        if (v != 0.0f) atomicAdd(&WX[(long)b * P_ + t], v);
    }
}

// ---------------------------------------------------------------------------
// WMMA f32 16x16x4 tile GEMM helper.
// A fragment: lane l holds row M = l&15; VGPR j holds K = (l>>4)*2 + j.
// B fragment: lane l holds col N = l&15; VGPR j holds K = (l>>4)*2 + j.
// C/D: VGPR r -> M = r + (l>>4)*8, N = l&15.
// ---------------------------------------------------------------------------
__device__ __forceinline__ v8f wmma_tile_f32(
    const float* __restrict__ A, int lda,
    const float* __restrict__ Bm, int ldb,
    int K, int mt, int nt, int lane)
{
    const int ml   = lane & 15;
    const int koff = (lane >> 4) * 2;
    const float* arow = A + (mt * 16 + ml) * lda;
    const float* bcol = Bm + nt * 16 + ml;
    v8f acc = {0.f, 0.f, 0.f, 0.f, 0.f, 0.f, 0.f, 0.f};
    for (int k = 0; k < K; k += 4) {
        v2f av, bv;
        av.x = arow[k + koff];
        av.y = arow[k + koff + 1];
        bv.x = bcol[(k + koff) * ldb];
        bv.y = bcol[(k + koff + 1) * ldb];
        acc = __builtin_amdgcn_wmma_f32_16x16x4_f32(
            false, av, false, bv, (short)0, acc, false, false);
    }
    return acc;
}

// ---------------------------------------------------------------------------
// Stage 2: fused head (BN0+CancelOut, 3x WMMA GEMM + bias/ReLU/BN, softmax).
// Single 512-thread block (16 full waves, EXEC always all-ones per wave).
// ---------------------------------------------------------------------------
__global__ void __launch_bounds__(512) spw_head(
    const float* __restrict__ WX,
    const float* __restrict__ co_w,
    const float* __restrict__ bn0_g, const float* __restrict__ bn0_b,
    const float* __restrict__ W1, const float* __restrict__ b1,
    const float* __restrict__ bn1_g, const float* __restrict__ bn1_b,
    const float* __restrict__ W2, const float* __restrict__ b2,
    const float* __restrict__ bn2_g, const float* __restrict__ bn2_b,
    const float* __restrict__ Wo, const float* __restrict__ bo,
    float* __restrict__ Zout,      // d_out + 2048 (64x128)
    float* __restrict__ yout,      // d_out       (64x32)
    float* __restrict__ h1,        // ws (64x256)
    float* __restrict__ h2,        // ws (64x128)
    float* __restrict__ logits)    // ws (64x32)
{
    __shared__ float sA[H1_];  // folded per-column scale
    __shared__ float sC[H1_];  // folded per-column shift
    const int t    = threadIdx.x;
    const int lane = t & 31;
    const int wave = t >> 5;
    const float invB = 1.0f / (float)B_;

    // ---- BN0 over relu(WX), folded with CancelOut sigmoid ----
    if (t < P_) {
        float s = 0.f, s2 = 0.f;
        for (int r = 0; r < B_; ++r) {
            float v = fmaxf(WX[r * P_ + t], 0.f);
            s += v; s2 += v * v;
        }
        float m   = s * invB;
        float var = s2 * invB - m * m;
        float rg  = rsqrtf(var + 1e-5f) * bn0_g[t];
        float sg  = 1.0f / (1.0f + __expf(-co_w[t]));
        sA[t] = rg * sg;
        sC[t] = (bn0_b[t] - m * rg) * sg;
    }
    __syncthreads();
    for (int e = t; e < B_ * P_; e += 512) {
        int cp = e & (P_ - 1);
        float v = fmaxf(WX[e], 0.f);
        Zout[e] = v * sA[cp] + sC[cp];
    }
    __syncthreads();

    // ---- GEMM1: relu(Z @ W1 + b1) -> h1 [64,256]; 4x16 tiles, 4/wave ----
    for (int q = 0; q < 4; ++q) {
        int tile = wave * 4 + q;
        int mt = tile >> 4, nt = tile & 15;
        v8f acc = wmma_tile_f32(Zout, P_, W1, H1_, P_, mt, nt, lane);
        int ml = lane & 15, n = nt * 16 + ml, mb = mt * 16 + (lane >> 4) * 8;
        float bias = b1[n];
        #pragma unroll
        for (int r = 0; r < 8; ++r)
            h1[(mb + r) * H1_ + n] = fmaxf(acc[r] + bias, 0.f);
    }
    __syncthreads();
    if (t < H1_) {
        float s = 0.f, s2 = 0.f;
        for (int r = 0; r < B_; ++r) { float v = h1[r * H1_ + t]; s += v; s2 += v * v; }
        float m = s * invB, var = s2 * invB - m * m;
        float rg = rsqrtf(var + 1e-5f) * bn1_g[t];
        sA[t] = rg;
        sC[t] = bn1_b[t] - m * rg;
    }
    __syncthreads();
    for (int e = t; e < B_ * H1_; e += 512) {
        int cc = e & (H1_ - 1);
        h1[e] = h1[e] * sA[cc] + sC[cc];
    }
    __syncthreads();

    // ---- GEMM2: relu(h1 @ W2 + b2) -> h2 [64,128]; 4x8 tiles, 2/wave ----
    for (int q = 0; q < 2; ++q) {
        int tile = wave * 2 + q;
        int mt = tile >> 3, nt = tile & 7;
        v8f acc = wmma_tile_f32(h1, H1_, W2, H2_, H1_, mt, nt, lane);
        int ml = lane & 15, n = nt * 16 + ml, mb = mt * 16 + (lane >> 4) * 8;
        float bias = b2[n];
        #pragma unroll
        for (int r = 0; r < 8; ++r)
            h2[(mb + r) * H2_ + n] = fmaxf(acc[r] + bias, 0.f);
    }
    __syncthreads();
    if (t < H2_) {
        float s = 0.f, s2 = 0.f;
        for (int r = 0; r < B_; ++r) { float v = h2[r * H2_ + t]; s += v; s2 += v * v; }
        float m = s * invB, var = s2 * invB - m * m;
        float rg = rsqrtf(var + 1e-5f) * bn2_g[t];
        sA[t] = rg;
        sC[t] = bn2_b[t] - m * rg;
    }
    __syncthreads();
    for (int e = t; e < B_ * H2_; e += 512) {
        int cc = e & (H2_ - 1);
        h2[e] = h2[e] * sA[cc] + sC[cc];
    }
    __syncthreads();

    // ---- GEMM3: h2 @ Wo + bo -> logits [64,32]; 4x2 tiles, waves 0..7 ----
    if (wave < 8) {
        int tile = wave;
        int mt = tile >> 1, nt = tile & 1;
        v8f acc = wmma_tile_f32(h2, H2_, Wo, C_, H2_, mt, nt, lane);
        int ml = lane & 15, n = nt * 16 + ml, mb = mt * 16 + (lane >> 4) * 8;
        float bias = bo[n];
        #pragma unroll
        for (int r = 0; r < 8; ++r)
            logits[(mb + r) * C_ + n] = acc[r] + bias;
    }
    __syncthreads();

    // ---- softmax rows -> y ----
    if (t < B_) {
        const float* lr = logits + t * C_;
        float m = -3.0e38f;
        for (int c = 0; c < C_; ++c) m = fmaxf(m, lr[c]);
        float ex[C_];
        float s = 0.f;
        for (int c = 0; c < C_; ++c) { float e = __expf(lr[c] - m); ex[c] = e; s += e; }
        float inv = 1.0f / s;
        for (int c = 0; c < C_; ++c) yout[t * C_ + c] = ex[c] * inv;
    }
}

// ---------------------------------------------------------------------------
// Launch. Inputs (setup_inputs order):
//  0:x 1:w 2:co_w 3:bn0_g 4:bn0_b 5:W1 6:b1 7:bn1_g 8:bn1_b
//  9:W2 10:b2 11:bn2_g 12:bn2_b 13:Wo 14:bo 15:idx
// d_out: y (64*32) then Z (64*128).  ws: WX(8192) | h1(16384) | h2(8192) | logits(2048)
// ---------------------------------------------------------------------------
extern "C" void kernel_launch(void* const* d_in, const int* in_sizes, int n_in,
                              void* d_out, int out_size, void* d_ws, size_t ws_size,
                              hipStream_t stream) {
    const float* x     = (const float*)d_in[0];
    const float* w     = (const float*)d_in[1];
    const float* co_w  = (const float*)d_in[2];
    const float* bn0_g = (const float*)d_in[3];
    const float* bn0_b = (const float*)d_in[4];
    const float* W1    = (const float*)d_in[5];
    const float* b1    = (const float*)d_in[6];
    const float* bn1_g = (const float*)d_in[7];
    const float* bn1_b = (const float*)d_in[8];
    const float* W2    = (const float*)d_in[9];
    const float* b2    = (const float*)d_in[10];
    const float* bn2_g = (const float*)d_in[11];
    const float* bn2_b = (const float*)d_in[12];
    const float* Wo    = (const float*)d_in[13];
    const float* bo    = (const float*)d_in[14];
    const int*   idx   = (const int*)d_in[15];

    float* out    = (float*)d_out;
    float* WX     = (float*)d_ws;
    float* h1     = WX + B_ * P_;
    float* h2     = h1 + B_ * H1_;
    float* logits = h2 + B_ * H2_;

    const int N = in_sizes[0] / B_;   // 800000

    hipMemsetAsync(WX, 0, (size_t)B_ * P_ * sizeof(float), stream);

    const int elems_per_block = 256 * 64;
    dim3 g1((N + elems_per_block - 1) / elems_per_block, B_);
    spw_scatter<<<g1, 256, 0, stream>>>(x, w, idx, WX, N);

    spw_head<<<1, 512, 0, stream>>>(WX, co_w, bn0_g, bn0_b,
                                    W1, b1, bn1_g, bn1_b,
                                    W2, b2, bn2_g, bn2_b,
                                    Wo, bo,
                                    out + B_ * C_, out,
                                    h1, h2, logits);
}